// SigKernel_41291815584050
// MI455X (gfx1250) — compile-verified
//
#include <hip/hip_runtime.h>

typedef __attribute__((ext_vector_type(2))) float v2f;
typedef __attribute__((ext_vector_type(8))) float v8f;

// Problem constants (reference: X,Y are (32, 256, 8) float32)
constexpr int kL = 256;          // path length
constexpr int kD = 8;            // channels
constexpr int kM = kL - 1;       // increments per path = 255
constexpr int kN = kM + 1;       // 256: diagonal length and padded inc dim

// One workgroup (256 threads = 8 wave32) per (a,b) pair.
//  Phase 1: inc = dX[a] @ dY[b]^T into LDS via V_WMMA_F32_16X16X4_F32 (K=8 as 2x4).
//  Phase 2: Goursat wavefront, registers carry K[d-1][i] and K[d-2][i-1];
//           only ONE neighbor LDS read + ONE store + ONE barrier per diagonal,
//           inc prefetched into f1/f2 before the barrier.
__global__ __launch_bounds__(256) void sigkernel_pde(const float* __restrict__ X,
                                                     const float* __restrict__ Y,
                                                     float* __restrict__ out,
                                                     int by) {
  __shared__ float s_inc[kN * kN];     // 256 KB: inc padded to 256x256
  __shared__ float s_kb[2][kN + 1];    // ping-pong diagonal exchange, [0] = guard

  const int a    = blockIdx.x;
  const int b    = blockIdx.y;
  const int tid  = threadIdx.x;
  const int wave = tid >> 5;
  const int lane = tid & 31;
  const int half = lane >> 4;          // 0: lanes 0-15, 1: lanes 16-31
  const int l15  = lane & 15;

  const float* __restrict__ Xa = X + (size_t)a * kL * kD;
  const float* __restrict__ Yb = Y + (size_t)b * kL * kD;

  // ---------------- Phase 1: inc tiles via WMMA ----------------
  for (int t = 0; t < 32; ++t) {
    const int tile = wave * 32 + t;
    const int tm = tile >> 4;
    const int tn = tile & 15;

    // clamp to last valid increment row (254); padded row/col never used
    int mrow = tm * 16 + l15; mrow = (mrow > kM - 1) ? (kM - 1) : mrow;
    int nrow = tn * 16 + l15; nrow = (nrow > kM - 1) ? (kM - 1) : nrow;
    const float* x0 = Xa + mrow * kD;
    const float* y0 = Yb + nrow * kD;

    v8f acc = {};
#pragma unroll
    for (int kk = 0; kk < 2; ++kk) {
      const int k0 = half * 2 + kk * 4;
      v2f Av, Bv;
      Av.x = x0[kD + k0]     - x0[k0];       // dX[m][k0]
      Av.y = x0[kD + k0 + 1] - x0[k0 + 1];   // dX[m][k0+1]
      Bv.x = y0[kD + k0]     - y0[k0];       // dY[n][k0]
      Bv.y = y0[kD + k0 + 1] - y0[k0 + 1];   // dY[n][k0+1]
      acc = __builtin_amdgcn_wmma_f32_16x16x4_f32(false, Av, false, Bv,
                                                  (short)0, acc, false, false);
    }

    const int rowbase = tm * 16 + half * 8;
    const int colbase = tn * 16 + l15;
#pragma unroll
    for (int r = 0; r < 8; ++r)
      s_inc[(rowbase + r) * kN + colbase] = acc[r];
  }

  if (tid == 0) { s_kb[0][0] = 0.0f; s_kb[1][0] = 0.0f; }   // neighbor guard
  __syncthreads();

  // ---------------- Phase 2: wavefront over antidiagonals ----------------
  const int  idx = tid;
  const bool is0 = (idx == 0);
  int ir = idx - 1; ir = (ir < 0) ? 0 : ((ir > kM - 1) ? (kM - 1) : ir);
  const float* __restrict__ incrow = &s_inc[ir * kN];   // one fixed row per thread

  // Register carries entering d=2:
  float km1s = (idx <= 1) ? 1.0f : 0.0f;                 // K[1][idx]
  float km1l = (idx == 1 || idx == 2) ? 1.0f : 0.0f;     // K[1][idx-1]
  float km2l = (idx == 1) ? 1.0f : 0.0f;                 // K[0][idx-1]

  // f1/f2 for d=2 (col index (d-1-idx) & 255; wrap only hits dead cells)
  float aa = incrow[(1 - idx) & (kN - 1)];
  float a2 = aa * aa;
  float f1 = fmaf(a2, (1.0f / 12.0f), fmaf(aa, 0.5f, 1.0f));
  float f2 = fmaf(a2, -(1.0f / 12.0f), 1.0f);

  // One diagonal step; pb is a compile-time ping-pong parity.
#define SIG_STEP(dd, pb)                                                     \
  {                                                                          \
    float kd = fmaf(km1s + km1l, f1, -(km2l * f2));                          \
    kd = (idx == (dd)) ? 1.0f : kd;          /* j == 0 boundary  */          \
    kd = is0 ? 1.0f : kd;                    /* i == 0 boundary  */          \
    s_kb[pb][idx + 1] = kd;                                                  \
    float aan = incrow[((dd) - idx) & (kN - 1)]; /* prefetch for dd+1 */     \
    __syncthreads();                                                         \
    float nbr = s_kb[pb][idx];               /* K[dd][idx-1]      */         \
    km2l = km1l; km1l = nbr; km1s = kd;                                      \
    float a2n = aan * aan;                                                   \
    f1 = fmaf(a2n, (1.0f / 12.0f), fmaf(aan, 0.5f, 1.0f));                   \
    f2 = fmaf(a2n, -(1.0f / 12.0f), 1.0f);                                   \
  }

  for (int d = 2; d <= 2 * kM; d += 2) {
    SIG_STEP(d, 0);
    if (d + 1 <= 2 * kM) SIG_STEP(d + 1, 1);   // uniform scalar condition
  }
#undef SIG_STEP

  // After the d = 2m step, km1s holds K[2m][idx]; answer is K[2m][m].
  if (idx == kM) out[a * by + b] = km1s;
}

extern "C" void kernel_launch(void* const* d_in, const int* in_sizes, int n_in,
                              void* d_out, int out_size, void* d_ws, size_t ws_size,
                              hipStream_t stream) {
  const float* X = (const float*)d_in[0];
  const float* Y = (const float*)d_in[1];
  float* out = (float*)d_out;

  const int bx = in_sizes[0] / (kL * kD);   // 32
  const int by = in_sizes[1] / (kL * kD);   // 32

  dim3 grid(bx, by);
  sigkernel_pde<<<grid, 256, 0, stream>>>(X, Y, out, by);
}